// Griddata_Cuda_28475633173083
// MI455X (gfx1250) — compile-verified
//
#include <hip/hip_runtime.h>

// Problem constants from the reference: B=4, C=3, H=W=1024.
constexpr int Bc = 4, Cc = 3, Hh = 1024, Ww = 1024;
constexpr int Nn = Hh * Ww;           // 1<<20
constexpr float kEps = 1e-8f;

// ---------------------------------------------------------------------------
// Zero-fill with float4 stores (keeps lines resident in L2 for atomic phase).
// ---------------------------------------------------------------------------
__global__ __launch_bounds__(256) void zero4_kernel(float4* __restrict__ p, int n4) {
  int i = blockIdx.x * blockDim.x + threadIdx.x;
  if (i < n4) p[i] = make_float4(0.f, 0.f, 0.f, 0.f);
}

// ---------------------------------------------------------------------------
// Forward splat: one thread per source pixel (b, n). Scatters den + 3 channel
// values to the 4 bilinear corners with native f32 global atomics.
// num == d_out (B*C*N floats), den == workspace (B*N floats).
// ---------------------------------------------------------------------------
__global__ __launch_bounds__(256) void splat_kernel(
    const float* __restrict__ im0,
    const float2* __restrict__ grid,
    float* __restrict__ num,
    float* __restrict__ den) {
  int gid = blockIdx.x * blockDim.x + threadIdx.x;   // exact: B*N threads
  int b = gid >> 20;                                  // N == 1<<20
  int n = gid & (Nn - 1);

  float2 g = grid[gid];                               // GLOBAL_LOAD_B64, coalesced
  float x0f = floorf(g.x);
  float y0f = floorf(g.y);
  float dx = g.x - x0f;
  float dy = g.y - y0f;
  int x0 = (int)x0f;
  int y0 = (int)y0f;

  const float* imb = im0 + (size_t)b * Cc * Nn + n;
  float v0 = imb[0];
  float v1 = imb[Nn];
  float v2 = imb[2 * Nn];

  float* numb = num + (size_t)b * Cc * Nn;
  float* denb = den + (size_t)b * Nn;

  float wx0 = 1.f - dx, wx1 = dx;
  float wy0 = 1.f - dy, wy1 = dy;
  float wgt[4] = {wx0 * wy0, wx1 * wy0, wx0 * wy1, wx1 * wy1};

#pragma unroll
  for (int k = 0; k < 4; ++k) {
    int xx = x0 + (k & 1);
    int yy = y0 + (k >> 1);
    // Reference zeroes the weight for out-of-range corners -> skipping is identical.
    if ((unsigned)xx < (unsigned)Ww && (unsigned)yy < (unsigned)Hh) {
      int idx = yy * Ww + xx;
      float w = wgt[k];
      unsafeAtomicAdd(denb + idx, w);               // global_atomic_add_f32 (no-return)
      unsafeAtomicAdd(numb + idx, w * v0);
      unsafeAtomicAdd(numb + idx + Nn, w * v1);
      unsafeAtomicAdd(numb + idx + 2 * Nn, w * v2);
    }
  }
}

// ---------------------------------------------------------------------------
// Normalize in place: out = num / max(den, EPS). float4 vectorized; den read
// once per 4 pixels and reused across the 3 channels.
// ---------------------------------------------------------------------------
__global__ __launch_bounds__(256) void normalize_kernel(
    float* __restrict__ out, const float* __restrict__ den) {
  int gid = blockIdx.x * blockDim.x + threadIdx.x;   // exact: B*N/4 threads
  constexpr int Q = Nn / 4;
  int b = gid / Q;
  int q = gid - b * Q;

  const float4 d = ((const float4*)(den + (size_t)b * Nn))[q];
  float4 m;
  m.x = 1.f / fmaxf(d.x, kEps);
  m.y = 1.f / fmaxf(d.y, kEps);
  m.z = 1.f / fmaxf(d.z, kEps);
  m.w = 1.f / fmaxf(d.w, kEps);

  float4* ob = (float4*)(out + (size_t)b * Cc * Nn);
#pragma unroll
  for (int c = 0; c < Cc; ++c) {
    float4 v = ob[(size_t)c * Q + q];
    v.x *= m.x; v.y *= m.y; v.z *= m.z; v.w *= m.w;
    ob[(size_t)c * Q + q] = v;
  }
}

// ---------------------------------------------------------------------------
// Harness entry point.
//   d_in[0] = im0  [B,C,H,W] f32
//   d_in[1] = grid [B,H,W,2] f32
//   d_out   = [B,C,H,W] f32 (also used as the num accumulator)
//   d_ws    = den accumulator, needs B*N*4 = 16 MB
// ---------------------------------------------------------------------------
extern "C" void kernel_launch(void* const* d_in, const int* in_sizes, int n_in,
                              void* d_out, int out_size, void* d_ws, size_t ws_size,
                              hipStream_t stream) {
  const float*  im0  = (const float*) d_in[0];
  const float2* grid = (const float2*)d_in[1];
  float* out = (float*)d_out;   // num accumulator, then final result
  float* den = (float*)d_ws;    // B*N floats

  const int n_out4 = Bc * Cc * Nn / 4;   // 3,145,728
  const int n_den4 = Bc * Nn / 4;        // 1,048,576
  zero4_kernel<<<(n_out4 + 255) / 256, 256, 0, stream>>>((float4*)out, n_out4);
  zero4_kernel<<<(n_den4 + 255) / 256, 256, 0, stream>>>((float4*)den, n_den4);

  const int n_src = Bc * Nn;             // 4,194,304 source pixels
  splat_kernel<<<n_src / 256, 256, 0, stream>>>(im0, grid, out, den);

  normalize_kernel<<<(Bc * Nn / 4) / 256, 256, 0, stream>>>(out, den);
}